// SPDEParameterGNN_36696200577142
// MI455X (gfx1250) — compile-verified
//
#include <hip/hip_runtime.h>

// ---------------------------------------------------------------------------
// 4-layer GCN for MI455X (gfx1250, wave32).
//   per layer: HT = H @ W  (fp32 WMMA 16x16x4, one wave = 16 nodes x F stripe)
//              acc[i] = HT[i]*dinv[i]^2 + b               (self loop + bias)
//              acc[dst] += HT[src] * dinv[src]*dinv[dst]  (edge scatter, f32 atomics)
//              next H = relu(acc)  (fused into consumer's A-load)
// All intermediates (~84 MB) are L2-resident (192 MB L2), so the scatter's
// random gathers/atomics stay on-chip; HBM sees each buffer roughly once.
// ---------------------------------------------------------------------------

typedef float v2f __attribute__((ext_vector_type(2)));
typedef float v8f __attribute__((ext_vector_type(8)));

#define TB 256
#define HID 64
#define IN_DIM 128

__device__ __forceinline__ void atomAddF(float* p, float v) {
  __hip_atomic_fetch_add(p, v, __ATOMIC_RELAXED, __HIP_MEMORY_SCOPE_AGENT);
}

// ---- degree / normalization precompute ------------------------------------

__global__ void fill_ones(float* __restrict__ p, int n) {
  int i = blockIdx.x * blockDim.x + threadIdx.x;
  if (i < n) p[i] = 1.0f;   // self-loop contributes 1 to every degree
}

__global__ void count_deg(const int* __restrict__ dst, float* deg, int E) {
  int e = blockIdx.x * blockDim.x + threadIdx.x;
  if (e < E) atomAddF(&deg[dst[e]], 1.0f);
}

__global__ void rsqrt_inplace(float* __restrict__ p, int n) {
  int i = blockIdx.x * blockDim.x + threadIdx.x;
  if (i < n) p[i] = rsqrtf(p[i]);   // deg >= 1 always (self loops)
}

__global__ void make_norm(const int* __restrict__ src, const int* __restrict__ dst,
                          const float* __restrict__ dinv, float* __restrict__ norm, int E) {
  int e = blockIdx.x * blockDim.x + threadIdx.x;
  if (e < E) norm[e] = dinv[src[e]] * dinv[dst[e]];
}

// ---- fp32 WMMA GEMM: HT[node, :] = (relu?)H[node, :] @ W -------------------
// One wave computes a full 16-node x F output stripe (F/16 accumulators),
// so each A fragment is loaded once and feeds F/16 WMMAs.
// A fragment (16x4 f32): half h = lane>>4, l = lane&15
//   a[r] = H[node0+l][k + 2h + r],  r = 0,1
// B fragment (4x16 f32):
//   b[r] = W[(k + 2h + r)*F + 16*t + l]
// D (16x16 f32): d[r] -> row = 8h + r, col = l
template <int K, int F, bool RELU>
__global__ __launch_bounds__(32)
void gemm_wmma_f32(const float* __restrict__ H, const float* __restrict__ W,
                   float* __restrict__ HT) {
  constexpr int NT = F / 16;            // column tiles per wave
  const int lane = threadIdx.x;
  const int half = lane >> 4;
  const int l    = lane & 15;
  const int node0 = blockIdx.x << 4;

  const float* __restrict__ arow = H + (size_t)(node0 + l) * K;

  v8f c[NT];
#pragma unroll
  for (int t = 0; t < NT; ++t) c[t] = (v8f){0.f, 0.f, 0.f, 0.f, 0.f, 0.f, 0.f, 0.f};

  for (int k = 0; k < K; k += 4) {
    const int ka = k + (half << 1);
    float a0 = arow[ka];
    float a1 = arow[ka + 1];
    if (RELU) { a0 = fmaxf(a0, 0.f); a1 = fmaxf(a1, 0.f); }
    v2f a; a.x = a0; a.y = a1;
#pragma unroll
    for (int t = 0; t < NT; ++t) {
      v2f b;
      b.x = W[ka * F + t * 16 + l];
      b.y = W[(ka + 1) * F + t * 16 + l];
      c[t] = __builtin_amdgcn_wmma_f32_16x16x4_f32(
          /*neg_a=*/false, a, /*neg_b=*/false, b,
          /*c_mod=*/(short)0, c[t], /*reuse_a=*/false, /*reuse_b=*/false);
    }
  }

#pragma unroll
  for (int t = 0; t < NT; ++t) {
    float* __restrict__ drow = HT + (size_t)(node0 + half * 8) * F + t * 16 + l;
#pragma unroll
    for (int r = 0; r < 8; ++r) drow[(size_t)r * F] = c[t][r];
  }
}

// ---- accumulator init: self-loop term + bias ------------------------------

__global__ void init_acc(const float* __restrict__ ht, const float* __restrict__ dinv,
                         const float* __restrict__ b, float* __restrict__ acc,
                         int n, int F) {
  int t = blockIdx.x * blockDim.x + threadIdx.x;
  if (t >= n * F) return;
  int i = t / F;
  int f = t - i * F;
  float s = dinv[i];
  acc[t] = ht[t] * (s * s) + b[f];
}

// ---- edge scatter: acc[dst] += ht[src] * norm (float4 gather, f32 atomics) -
// 16 consecutive lanes handle one edge (F=64) -> coalesced float4 row gather.

__global__ void edge_scatter(const int* __restrict__ src, const int* __restrict__ dst,
                             const float* __restrict__ norm, const float* __restrict__ ht,
                             float* acc, int E, int F) {
  const int chunks = F >> 2;                  // float4 chunks per row
  int t = blockIdx.x * blockDim.x + threadIdx.x;
  if (t >= E * chunks) return;
  int e = t / chunks;
  int c = t - e * chunks;
  int s = src[e], d = dst[e];
  float w = norm[e];
  const float4* __restrict__ row = reinterpret_cast<const float4*>(ht + (size_t)s * F);
  float4 v = row[c];
  float* out = acc + (size_t)d * F + (c << 2);
  atomAddF(out + 0, v.x * w);
  atomAddF(out + 1, v.y * w);
  atomAddF(out + 2, v.z * w);
  atomAddF(out + 3, v.w * w);
}

// ---- head: out = relu(h) @ Wh + bh, 32 -> 3 -------------------------------

__global__ void head3(const float* __restrict__ h, const float* __restrict__ Wh,
                      const float* __restrict__ bh, float* __restrict__ out, int n) {
  int i = blockIdx.x * blockDim.x + threadIdx.x;
  if (i >= n) return;
  float a0 = bh[0], a1 = bh[1], a2 = bh[2];
  const float* hr = h + (size_t)i * 32;
#pragma unroll
  for (int k = 0; k < 32; ++k) {
    float v = fmaxf(hr[k], 0.f);
    a0 += v * Wh[k * 3 + 0];
    a1 += v * Wh[k * 3 + 1];
    a2 += v * Wh[k * 3 + 2];
  }
  out[i * 3 + 0] = a0;
  out[i * 3 + 1] = a1;
  out[i * 3 + 2] = a2;
}

// ---------------------------------------------------------------------------

extern "C" void kernel_launch(void* const* d_in, const int* in_sizes, int n_in,
                              void* d_out, int out_size, void* d_ws, size_t ws_size,
                              hipStream_t stream) {
  const float* x  = (const float*)d_in[0];
  const int*   ei = (const int*)d_in[1];
  const float* W1 = (const float*)d_in[2];
  const float* b1 = (const float*)d_in[3];
  const float* W2 = (const float*)d_in[4];
  const float* b2 = (const float*)d_in[5];
  const float* W3 = (const float*)d_in[6];
  const float* b3 = (const float*)d_in[7];
  const float* W4 = (const float*)d_in[8];
  const float* b4 = (const float*)d_in[9];
  const float* Wh = (const float*)d_in[10];
  const float* bh = (const float*)d_in[11];
  float* out = (float*)d_out;

  const int N = in_sizes[0] / IN_DIM;   // 100000 (divisible by 16)
  const int E = in_sizes[1] / 2;        // 1600000
  const int* src = ei;                  // edge_index[0]
  const int* dst = ei + E;              // edge_index[1]

  // workspace layout (floats); all offsets 16B-aligned
  float* ws   = (float*)d_ws;
  float* dinv = ws;                               // N
  float* nrm  = dinv + N;                         // E
  float* ht   = nrm + E;                          // N*64
  float* hA   = ht + (size_t)N * HID;             // N*64
  float* hB   = hA + (size_t)N * HID;             // N*64

  const int nodeTiles = N / 16;

  // --- degree / normalization ---
  fill_ones<<<(N + TB - 1) / TB, TB, 0, stream>>>(dinv, N);
  count_deg<<<(E + TB - 1) / TB, TB, 0, stream>>>(dst, dinv, E);
  rsqrt_inplace<<<(N + TB - 1) / TB, TB, 0, stream>>>(dinv, N);
  make_norm<<<(E + TB - 1) / TB, TB, 0, stream>>>(src, dst, dinv, nrm, E);

  // --- post-GEMM aggregation (shared by all layers) ---
  auto aggregate = [&](const float* b, int F, float* accbuf) {
    int nf = N * F;
    init_acc<<<(nf + TB - 1) / TB, TB, 0, stream>>>(ht, dinv, b, accbuf, N, F);
    int tot = E * (F >> 2);
    edge_scatter<<<(tot + TB - 1) / TB, TB, 0, stream>>>(src, dst, nrm, ht, accbuf, E, F);
  };

  // L1: x[N,128] @ W1[128,64]
  gemm_wmma_f32<IN_DIM, HID, false><<<nodeTiles, 32, 0, stream>>>(x, W1, ht);
  aggregate(b1, HID, hA);
  // L2: relu(hA) @ W2[64,64]
  gemm_wmma_f32<HID, HID, true><<<nodeTiles, 32, 0, stream>>>(hA, W2, ht);
  aggregate(b2, HID, hB);
  // L3: relu(hB) @ W3[64,64]
  gemm_wmma_f32<HID, HID, true><<<nodeTiles, 32, 0, stream>>>(hB, W3, ht);
  aggregate(b3, HID, hA);
  // L4: relu(hA) @ W4[64,32]
  gemm_wmma_f32<HID, HID / 2, true><<<nodeTiles, 32, 0, stream>>>(hA, W4, ht);
  aggregate(b4, HID / 2, hB);

  // head: relu(hB) @ Wh[32,3] + bh
  head3<<<(N + TB - 1) / TB, TB, 0, stream>>>(hB, Wh, bh, out, N);
}